// MyModel_87522843559970
// MI455X (gfx1250) — compile-verified
//
#include <hip/hip_runtime.h>
#include <stdint.h>

typedef __attribute__((ext_vector_type(2))) float v2f;
typedef __attribute__((ext_vector_type(8))) float v8f;

#define B_   8
#define H_   224
#define W_   224
#define C_   96
#define KW   64
#define PAD  63          // dilated kernel extent = 2*(64-1)+1 = 127, SAME pad = 63
#define XT   16          // output x-tile (WMMA M)
#define YB   64          // output rows per workgroup
#define NYIN (YB + 2*PAD)   // 190 staged input rows
#define SCOLS 142           // staged input cols: XT-1 + 2*(KW-1) + 1
#define SSTRIDE 144         // padded LDS row stride

// Generic (flat) pointer to LDS: low 32 bits are the LDS byte offset (ISA 10.2).
__device__ __forceinline__ unsigned lds_off(const void* p) {
    return (unsigned)(uintptr_t)p;
}

__global__ __launch_bounds__(256)
void dwconv_wmma_f32(const float* __restrict__ in, const float* __restrict__ kw,
                     float* __restrict__ out)
{
    extern __shared__ float smem[];
    float* s_in  = smem;                       // NYIN * SSTRIDE floats
    float* s_w   = s_in + NYIN * SSTRIDE;      // KW*KW floats (this channel's kernel)
    float* s_acc = s_w + KW * KW;              // YB * XT output accumulator

    const int bc  = blockIdx.x;
    const int b   = bc / C_;
    const int c   = bc % C_;
    const int Y0  = blockIdx.y * YB;
    const int X0  = blockIdx.z * XT;
    const int tid = threadIdx.x;

    // --- zero the LDS output accumulator (ds_store) ---
    for (int i = tid; i < YB * XT; i += 256) s_acc[i] = 0.0f;

    // --- async-stage this channel's 64x64 kernel: global -> LDS, no VGPR round-trip ---
    for (int i = tid; i < KW * KW; i += 256) {
        unsigned l = lds_off(&s_w[i]);
        const float* g = &kw[(size_t)i * C_ + c];
        asm volatile("global_load_async_to_lds_b32 %0, %1, off"
                     :: "v"(l), "v"(g) : "memory");
    }

    // --- async-stage zero-padded input patch: rows [Y0-63, Y0+126], cols [X0-63, X0+78] ---
    for (int i = tid; i < NYIN * SCOLS; i += 256) {
        int r  = i / SCOLS;
        int cc = i - r * SCOLS;
        int y  = Y0 - PAD + r;
        int x  = X0 - PAD + cc;
        if ((unsigned)y < (unsigned)H_ && (unsigned)x < (unsigned)W_) {
            unsigned l = lds_off(&s_in[r * SSTRIDE + cc]);
            const float* g = &in[(((size_t)b * H_ + y) * W_ + x) * C_ + c];
            asm volatile("global_load_async_to_lds_b32 %0, %1, off"
                         :: "v"(l), "v"(g) : "memory");
        } else {
            s_in[r * SSTRIDE + cc] = 0.0f;     // padding via ds_store (disjoint addrs)
        }
    }

    asm volatile("s_wait_asynccnt 0" ::: "memory");  // drain ASYNCcnt before the barrier
    __syncthreads();

    const int lane  = tid & 31;
    const int wave  = tid >> 5;
    const int m     = lane & 15;   // A: M index; B: N index; D: N index
    const int khalf = lane >> 4;   // A/B: which K half; D: +8 on M

    // Work units: (staged input row r) x (ky chunk of 16).  8 waves round-robin.
    for (int u = wave; u < NYIN * 4; u += 8) {
        const int r    = u >> 2;
        const int kyc  = u & 3;
        const int y_in = Y0 - PAD + r;
        if ((unsigned)y_in >= (unsigned)H_) continue;   // zero rows contribute nothing

        const int kyBase = kyc * 16;
        // output rows touched by this chunk: y_in + 63 - 2*(kyBase .. kyBase+15)
        const int yo_hi = y_in + PAD - 2 * kyBase;
        const int yo_lo = yo_hi - 30;
        if (yo_hi < Y0 || yo_lo >= Y0 + YB) continue;   // misses this tile entirely

        v8f acc = {};
        const float* srow = s_in + r * SSTRIDE;

        // Chain 16 x (16x16x4 f32) WMMAs: full kx = 0..63 contraction in registers.
        #pragma unroll
        for (int j = 0; j < 16; ++j) {
            // A[m, k] = in[y_in, X0 + m + 2*kx - 63]; staged col = m + 2*kx
            const int colBase = m + 8 * j + 4 * khalf;  // kx = 4j + 2*khalf (+1 for .y)
            v2f a;
            a.x = srow[colBase];
            a.y = srow[colBase + 2];

            // B[k, n] = w[kyBase + n, kx], n = m
            const int kx = 4 * j + 2 * khalf;
            v2f bmat;
            bmat.x = s_w[(kyBase + m) * KW + kx];
            bmat.y = s_w[(kyBase + m) * KW + kx + 1];

            acc = __builtin_amdgcn_wmma_f32_16x16x4_f32(
                /*neg_a=*/false, a, /*neg_b=*/false, bmat,
                /*c_mod=*/(short)0, acc, /*reuse_a=*/false, /*reuse_b=*/false);
        }

        // Scatter D columns: lane holds D[v + 8*khalf, m] in acc[v]
        #pragma unroll
        for (int vv = 0; vv < 8; ++vv) {
            const int md  = vv + 8 * khalf;                 // x within tile
            const int yo  = y_in + PAD - 2 * (kyBase + m);  // output row for column n=m
            const int row = yo - Y0;
            if ((unsigned)row < (unsigned)YB)
                atomicAdd(&s_acc[row * XT + md], acc[vv]);  // ds_add_f32
        }
    }
    __syncthreads();

    // --- write the output tile (NHWC) asynchronously: LDS -> global ---
    for (int i = tid; i < YB * XT; i += 256) {
        int row = i / XT;
        int xc  = i - row * XT;
        int y   = Y0 + row;
        if (y < H_) {
            unsigned l = lds_off(&s_acc[i]);
            float* g = &out[(((size_t)b * H_ + y) * W_ + X0 + xc) * C_ + c];
            asm volatile("global_store_async_from_lds_b32 %0, %1, off"
                         :: "v"(g), "v"(l) : "memory");
        }
    }
    asm volatile("s_wait_asynccnt 0" ::: "memory");  // drain before wave exit
}

extern "C" void kernel_launch(void* const* d_in, const int* in_sizes, int n_in,
                              void* d_out, int out_size, void* d_ws, size_t ws_size,
                              hipStream_t stream) {
    (void)in_sizes; (void)n_in; (void)d_ws; (void)ws_size; (void)out_size;
    const float* in = (const float*)d_in[0];
    const float* kw = (const float*)d_in[1];
    float* out      = (float*)d_out;

    dim3 grid(B_ * C_, (H_ + YB - 1) / YB, W_ / XT);   // 768 x 4 x 14
    size_t shmem = (size_t)(NYIN * SSTRIDE + KW * KW + YB * XT) * sizeof(float);
    dwconv_wmma_f32<<<grid, 256, shmem, stream>>>(in, kw, out);
}